// ASAP_89318139887646
// MI455X (gfx1250) — compile-verified
//
#include <hip/hip_runtime.h>
#include <hip/hip_bf16.h>

// ---------------------------------------------------------------------------
// CDNA5 (gfx1250) GCN forward. fp32 WMMA (v_wmma_f32_16x16x4_f32) for all
// node-feature GEMMs; BOTH operands staged into LDS by the Tensor Data Mover
// (2-D strided descriptor + LDS row padding for A, 1-D for B), inner loop is
// pure ds_load + v_wmma. Atomic-f32 edge scatter dominates the memory
// roofline (~6 GB @ 23.3 TB/s), so fp32 matrix math is the right precision.
// ---------------------------------------------------------------------------

typedef __attribute__((ext_vector_type(2))) float v2f;
typedef __attribute__((ext_vector_type(8))) float v8f;
typedef __attribute__((ext_vector_type(4))) unsigned int v4u;
typedef __attribute__((ext_vector_type(4))) int v4i;
typedef __attribute__((ext_vector_type(8))) int v8i;

#define GN 100000
#define GE 1600000
#define GG 1000
#define GIN 64
#define GHID 128
#define GEX 32
#define GOUT 10
#define GKC 32            // K-chunk; all K used (64,128,416) are multiples of 32
#define GRSA (GKC + 2)    // padded LDS row stride for A (TDM pad: 2 dwords/row)

#if defined(__HIP_DEVICE_COMPILE__) && __has_builtin(__builtin_amdgcn_tensor_load_to_lds)
#define ATHENA_TDM 1
#else
#define ATHENA_TDM 0
#endif

// ---------------------------------------------------------------- utilities
__global__ void k_zero(float* __restrict__ p, size_t n) {
  size_t i = (size_t)blockIdx.x * blockDim.x + threadIdx.x;
  if (i < n) p[i] = 0.0f;
}

__global__ void k_deg_accum(const int* __restrict__ dst, float* __restrict__ deg) {
  int e = blockIdx.x * blockDim.x + threadIdx.x;
  if (e < GE) atomicAdd(&deg[dst[e]], 1.0f);
}

__global__ void k_deg_finalize(float* __restrict__ dinv) {
  int i = blockIdx.x * blockDim.x + threadIdx.x;
  if (i < GN) dinv[i] = rsqrtf(dinv[i] + 1.0f);  // (A+I) degree
}

__global__ void k_count_nodes(const int* __restrict__ batch, float* __restrict__ cnt) {
  int i = blockIdx.x * blockDim.x + threadIdx.x;
  if (i < GN) atomicAdd(&cnt[batch[i]], 1.0f);
}

// ------------------------------------------------------- WMMA fp32 GEMM
// C[M x 128] = A[M x K] @ B[K x 128]   (optionally += ), K % 32 == 0.
// block = 256 threads = 8 waves; block covers 128 rows x 128 cols.
// wave w -> column tile w; 8 row tiles accumulated in 8 x v8f per lane.
// Per K-chunk: TDM loads A tile (128 x 32, global row stride K, LDS rows
// padded +2 dwords) and B tile (32 x 128, contiguous) into LDS; inner loop
// is ds_load_b64 x8 (A) + dual-offset ds load (B) + v_wmma x8.
__global__ __launch_bounds__(256) void k_gemm128_wmma(
    const float* __restrict__ A, const float* __restrict__ B,
    float* __restrict__ C, int M, int K, int accumulate) {
  __shared__ __align__(16) float ldsA[128 * GRSA];   // 17408 B
  __shared__ __align__(16) float ldsB[GKC * 128];    // 16384 B

  const int lane = threadIdx.x & 31;
  const int wave = threadIdx.x >> 5;
  const int hi = lane >> 4;     // 0/1 : selects K pair within quad
  const int ln = lane & 15;     // row (A) / col (B,C,D) within tile
  const int m0 = blockIdx.x * 128;
  const int col = wave * 16 + ln;

  const v8f vz = {0.f, 0.f, 0.f, 0.f, 0.f, 0.f, 0.f, 0.f};
  v8f acc[8];
#pragma unroll
  for (int rt = 0; rt < 8; ++rt) acc[rt] = vz;

  for (int kc = 0; kc < K; kc += GKC) {
    __syncthreads();  // prior chunk's LDS reads complete
#if ATHENA_TDM
    if (threadIdx.x < 32) {
      const unsigned ldsA_off = (unsigned)(size_t)(void*)&ldsA[0];
      const unsigned ldsB_off = (unsigned)(size_t)(void*)&ldsB[0];
      v4i gz = {0, 0, 0, 0};
      // ---- descriptor 1: A tile, 2-D (tile 32 x 128, tensor stride K),
      //      LDS pad: +2 dwords every 32 dwords -> row stride GRSA.
      {
        unsigned long long ga =
            (unsigned long long)(const void*)(A + (size_t)m0 * K + kc);
        unsigned rows = (unsigned)(M - m0);  // OOB rows read back as zero
        v4u g0;
        g0.x = 1u;                                   // count=1 user descriptor
        g0.y = ldsA_off;                             // lds_addr
        g0.z = (unsigned)ga;                         // global_addr[31:0]
        g0.w = (unsigned)((ga >> 32) & 0x01FFFFFFu) | (2u << 30);
        v8i g1;
        g1[0] = (int)((2u << 16) |                   // data_size = 4 B
                      (1u << 20) |                   // pad_enable
                      (4u << 22) |                   // pad_interval: 32 dwords
                      (1u << 25));                   // pad_amount: 2 dwords
        g1[1] = (int)(((unsigned)K & 0xFFFFu) << 16);          // tensor_dim0 lo
        g1[2] = (int)(((unsigned)K >> 16) | ((rows & 0xFFFFu) << 16));
        g1[3] = (int)((rows >> 16) | ((unsigned)GKC << 16));   // tile_dim0=32
        g1[4] = 128;                                           // tile_dim1=128
        g1[5] = (int)K;                                        // dim0_stride
        g1[6] = 0;
        g1[7] = 0;
#if __clang_major__ >= 23
        v8i gz8 = {0, 0, 0, 0, 0, 0, 0, 0};
        __builtin_amdgcn_tensor_load_to_lds(g0, g1, gz, gz, gz8, 0);
#else
        __builtin_amdgcn_tensor_load_to_lds(g0, g1, gz, gz, 0);
#endif
      }
      // ---- descriptor 2: B chunk, 1-D copy of 32*128 contiguous floats
      {
        unsigned long long ga =
            (unsigned long long)(const void*)(B + (size_t)kc * 128);
        const unsigned nelem = GKC * 128;  // 4096
        v4u g0;
        g0.x = 1u;
        g0.y = ldsB_off;
        g0.z = (unsigned)ga;
        g0.w = (unsigned)((ga >> 32) & 0x01FFFFFFu) | (2u << 30);
        v8i g1;
        g1[0] = (int)(2u << 16);                     // data_size = 4 B, no pad
        g1[1] = (int)((nelem & 0xFFFFu) << 16);      // tensor_dim0[15:0]
        g1[2] = (int)((nelem >> 16) | (1u << 16));   // tensor_dim1 = 1
        g1[3] = (int)(nelem << 16);                  // tile_dim0 = 4096
        g1[4] = 0;
        g1[5] = (int)nelem;                          // tensor_dim0_stride
        g1[6] = 0;
        g1[7] = 0;
#if __clang_major__ >= 23
        v8i gz8 = {0, 0, 0, 0, 0, 0, 0, 0};
        __builtin_amdgcn_tensor_load_to_lds(g0, g1, gz, gz, gz8, 0);
#else
        __builtin_amdgcn_tensor_load_to_lds(g0, g1, gz, gz, 0);
#endif
      }
#if __has_builtin(__builtin_amdgcn_s_wait_tensorcnt)
      __builtin_amdgcn_s_wait_tensorcnt(0);
#else
      asm volatile("s_wait_tensorcnt 0x0" ::: "memory");
#endif
    }
    {  // compiler barrier: LDS was written behind the compiler's back
      float* lp = &ldsA[0];
      asm volatile("" : "+r"(lp)::"memory");
    }
#else
    {  // fallback: cooperative staging (same padded layout)
      for (int i = threadIdx.x; i < 128 * GKC; i += 256) {
        int r = i >> 5;           // / GKC
        int c = i & (GKC - 1);    // % GKC
        int grow = m0 + r;
        grow = grow < M ? grow : M - 1;
        ldsA[r * GRSA + c] = A[(size_t)grow * K + kc + c];
      }
      const float4* Bv = (const float4*)(B + (size_t)kc * 128);
      float4* Lv = (float4*)ldsB;
      for (int i = threadIdx.x; i < GKC * 32; i += 256) Lv[i] = Bv[i];
    }
#endif
    __syncthreads();

#pragma unroll
    for (int kk = 0; kk < GKC; kk += 4) {
      // B fragment (4x16): lane ln = col, hi selects K pair, vgpr j = K lsb
      v2f b;
      b.x = ldsB[(kk + 2 * hi) * 128 + col];
      b.y = ldsB[(kk + 2 * hi + 1) * 128 + col];
      // A fragments for all 8 row tiles: batched ds_load_b64, one wait
      v2f a[8];
#pragma unroll
      for (int rt = 0; rt < 8; ++rt)
        a[rt] = *(const v2f*)&ldsA[(rt * 16 + ln) * GRSA + kk + 2 * hi];
#pragma unroll
      for (int rt = 0; rt < 8; ++rt)
        acc[rt] = __builtin_amdgcn_wmma_f32_16x16x4_f32(
            false, a[rt], false, b, (short)0, acc[rt], false, false);
    }
  }

#pragma unroll
  for (int rt = 0; rt < 8; ++rt) {
#pragma unroll
    for (int v = 0; v < 8; ++v) {
      int row = m0 + rt * 16 + 8 * hi + v;  // C/D layout: M = v + 8*(lane/16)
      if (row < M) {
        size_t o = (size_t)row * 128 + col;
        float val = acc[rt][v];
        if (accumulate) val += C[o];
        C[o] = val;
      }
    }
  }
}

// ----------------------------------------------------- edge scatter-add
// agg[dst] += xw[src] * dinv[src]*dinv[dst]; one thread per (edge, feature)
__global__ void k_scatter(const float* __restrict__ xw, const int* __restrict__ src,
                          const int* __restrict__ dst, const float* __restrict__ dinv,
                          float* __restrict__ agg) {
  size_t idx = (size_t)blockIdx.x * blockDim.x + threadIdx.x;
  if (idx >= (size_t)GE * GHID) return;
  int e = (int)(idx >> 7);
  int f = (int)(idx & 127);
  int s = src[e], d = dst[e];
  float c = dinv[s] * dinv[d];
  atomicAdd(&agg[(size_t)d * GHID + f], xw[(size_t)s * GHID + f] * c);
}

// y = agg + xw*dinv^2 + b (in place in agg); accumulate column sum / sumsq
__global__ void k_conv_finalize(float* __restrict__ y, const float* __restrict__ xw,
                                const float* __restrict__ dinv, const float* __restrict__ b,
                                float* __restrict__ csum, float* __restrict__ csumsq) {
  int gtid = blockIdx.x * blockDim.x + threadIdx.x;
  int f = gtid & 127;
  int row0 = gtid >> 7;
  int rstride = (gridDim.x * blockDim.x) >> 7;
  float bf = b[f];
  float s = 0.f, s2 = 0.f;
  for (int i = row0; i < GN; i += rstride) {
    float di = dinv[i];
    size_t o = (size_t)i * GHID + f;
    float v = y[o] + xw[o] * di * di + bf;
    y[o] = v;
    s += v;
    s2 += v * v;
  }
  atomicAdd(&csum[f], s);
  atomicAdd(&csumsq[f], s2);
}

// h = relu((y - mu) * rsqrt(var+eps) * g + be)   (in place)
__global__ void k_bn_relu(float* __restrict__ y, const float* __restrict__ csum,
                          const float* __restrict__ csumsq, const float* __restrict__ g,
                          const float* __restrict__ be) {
  size_t idx = (size_t)blockIdx.x * blockDim.x + threadIdx.x;
  if (idx >= (size_t)GN * GHID) return;
  int f = (int)(idx & 127);
  float invN = 1.0f / (float)GN;
  float mu = csum[f] * invN;
  float var = csumsq[f] * invN - mu * mu;
  float v = (y[idx] - mu) * rsqrtf(var + 1e-5f) * g[f] + be[f];
  y[idx] = v > 0.f ? v : 0.f;
}

// z = sigmoid(t + bi + bo); hn = z*out + (1-z)*in; mean-pool accumulate
__global__ void k_gate_pool(const float* __restrict__ t, const float* __restrict__ outx,
                            const float* __restrict__ inx, const float* __restrict__ bi,
                            const float* __restrict__ bo, float* __restrict__ hn,
                            float* __restrict__ pool, const int* __restrict__ batch) {
  size_t idx = (size_t)blockIdx.x * blockDim.x + threadIdx.x;
  if (idx >= (size_t)GN * GHID) return;
  int f = (int)(idx & 127);
  int i = (int)(idx >> 7);
  float tt = t[idx] + bi[f] + bo[f];
  float z = 1.0f / (1.0f + __expf(-tt));
  float v = z * outx[idx] + (1.0f - z) * inx[idx];
  hn[idx] = v;
  atomicAdd(&pool[(size_t)batch[i] * GHID + f], v);
}

// hg = [p1/cnt | p2/cnt | p3/cnt | eFeature]   (G x 416)
__global__ void k_build_hg(const float* __restrict__ p1, const float* __restrict__ p2,
                           const float* __restrict__ p3, const float* __restrict__ cnt,
                           const float* __restrict__ eF, float* __restrict__ hg) {
  int idx = blockIdx.x * blockDim.x + threadIdx.x;
  const int W = 3 * GHID + GEX;  // 416
  if (idx >= GG * W) return;
  int g = idx / W;
  int f = idx - g * W;
  float v;
  if (f < 3 * GHID) {
    int which = f >> 7;
    int ff = f & 127;
    const float* p = (which == 0) ? p1 : (which == 1) ? p2 : p3;
    float c = cnt[g];
    v = p[(size_t)g * GHID + ff] / (c > 1.0f ? c : 1.0f);
  } else {
    v = eF[(size_t)g * GEX + (f - 3 * GHID)];
  }
  hg[idx] = v;
}

__global__ void k_relu_bias(float* __restrict__ h, const float* __restrict__ b) {
  int idx = blockIdx.x * blockDim.x + threadIdx.x;
  if (idx >= GG * GHID) return;
  float v = h[idx] + b[idx & 127];
  h[idx] = v > 0.f ? v : 0.f;
}

__global__ void k_fc3(const float* __restrict__ h, const float* __restrict__ W,
                      const float* __restrict__ b, float* __restrict__ out) {
  int idx = blockIdx.x * blockDim.x + threadIdx.x;
  if (idx >= GG * GOUT) return;
  int g = idx / GOUT;
  int o = idx - g * GOUT;
  float s = b[o];
  for (int k = 0; k < GHID; ++k) s += h[(size_t)g * GHID + k] * W[k * GOUT + o];
  out[idx] = s;
}

// ---------------------------------------------------------------- host side
static inline int cdiv(long long a, long long b) { return (int)((a + b - 1) / b); }

extern "C" void kernel_launch(void* const* d_in, const int* in_sizes, int n_in,
                              void* d_out, int out_size, void* d_ws, size_t ws_size,
                              hipStream_t stream) {
  // inputs in setup_inputs() dict order
  const float* x = (const float*)d_in[0];
  const float* eF = (const float*)d_in[1];
  const int* ei = (const int*)d_in[2];
  const int* batch = (const int*)d_in[3];
  const float* W1 = (const float*)d_in[4];
  const float* b1 = (const float*)d_in[5];
  const float* W2 = (const float*)d_in[6];
  const float* b2 = (const float*)d_in[7];
  const float* W3 = (const float*)d_in[8];
  const float* b3 = (const float*)d_in[9];
  const float* g1 = (const float*)d_in[10];
  const float* be1 = (const float*)d_in[11];
  const float* g2 = (const float*)d_in[12];
  const float* be2 = (const float*)d_in[13];
  const float* g3 = (const float*)d_in[14];
  const float* be3 = (const float*)d_in[15];
  const float* s1_Wp = (const float*)d_in[16];
  const float* s1_Wi = (const float*)d_in[17];
  const float* s1_bi = (const float*)d_in[18];
  const float* s1_Wo = (const float*)d_in[19];
  const float* s1_bo = (const float*)d_in[20];
  const float* s2_Wi = (const float*)d_in[21];
  const float* s2_bi = (const float*)d_in[22];
  const float* s2_Wo = (const float*)d_in[23];
  const float* s2_bo = (const float*)d_in[24];
  const float* s3_Wi = (const float*)d_in[25];
  const float* s3_bi = (const float*)d_in[26];
  const float* s3_Wo = (const float*)d_in[27];
  const float* s3_bo = (const float*)d_in[28];
  const float* fc1_W = (const float*)d_in[29];
  const float* fc1_b = (const float*)d_in[30];
  const float* fc3_W = (const float*)d_in[31];
  const float* fc3_b = (const float*)d_in[32];

  const int* src = ei;
  const int* dst = ei + GE;

  // workspace carve
  float* ws = (float*)d_ws;
  size_t off = 0;
  float* XW = ws + off;   off += (size_t)GN * GHID;  // xw / gate pre-activation t
  float* AGG = ws + off;  off += (size_t)GN * GHID;  // agg -> y -> h (layer local)
  float* BUF0 = ws + off; off += (size_t)GN * GHID;
  float* BUF1 = ws + off; off += (size_t)GN * GHID;
  float* dinv = ws + off; off += GN;
  float* csum = ws + off; off += GHID;
  float* csumsq = ws + off; off += GHID;
  float* p1 = ws + off;   off += (size_t)GG * GHID;
  float* p2 = ws + off;   off += (size_t)GG * GHID;
  float* p3 = ws + off;   off += (size_t)GG * GHID;
  float* cnt = ws + off;  off += GG;
  float* hg = ws + off;   off += (size_t)GG * (3 * GHID + GEX);
  float* hid2 = ws + off; off += (size_t)GG * GHID;

  const size_t NH = (size_t)GN * GHID;
  const size_t EH = (size_t)GE * GHID;

  auto zero = [&](float* p, size_t n) {
    k_zero<<<cdiv((long long)n, 256), 256, 0, stream>>>(p, n);
  };
  auto gemm = [&](const float* A, const float* B, float* C, int M, int K, int acc) {
    k_gemm128_wmma<<<cdiv(M, 128), 256, 0, stream>>>(A, B, C, M, K, acc);
  };

  // degrees / graph counts / pool accumulators
  zero(dinv, GN);
  zero(cnt, GG);
  zero(p1, (size_t)3 * GG * GHID);  // p1,p2,p3 contiguous
  k_deg_accum<<<cdiv(GE, 256), 256, 0, stream>>>(dst, dinv);
  k_deg_finalize<<<cdiv(GN, 256), 256, 0, stream>>>(dinv);
  k_count_nodes<<<cdiv(GN, 256), 256, 0, stream>>>(batch, cnt);

  auto layer = [&](const float* in_feat, int in_dim, const float* W, const float* b,
                   const float* g, const float* be, const float* Wi, const float* bi,
                   const float* Wo, const float* bo, const float* skip_in,
                   float* h_out, float* pool) {
    gemm(in_feat, W, XW, GN, in_dim, 0);                           // xw
    zero(AGG, NH);
    k_scatter<<<cdiv((long long)EH, 256), 256, 0, stream>>>(XW, src, dst, dinv, AGG);
    zero(csum, GHID);
    zero(csumsq, GHID);
    k_conv_finalize<<<1024, 256, 0, stream>>>(AGG, XW, dinv, b, csum, csumsq);
    k_bn_relu<<<cdiv((long long)NH, 256), 256, 0, stream>>>(AGG, csum, csumsq, g, be);
    gemm(AGG, Wo, XW, GN, GHID, 0);                                // t  = h @ Wo
    gemm(skip_in, Wi, XW, GN, GHID, 1);                            // t += in @ Wi
    k_gate_pool<<<cdiv((long long)NH, 256), 256, 0, stream>>>(
        XW, AGG, skip_in, bi, bo, h_out, pool, batch);
  };

  // layer 1: skip input is projected x
  gemm(x, s1_Wp, BUF0, GN, GIN, 0);
  layer(x, GIN, W1, b1, g1, be1, s1_Wi, s1_bi, s1_Wo, s1_bo, BUF0, BUF1, p1);  // h1=BUF1
  // layer 2
  layer(BUF1, GHID, W2, b2, g2, be2, s2_Wi, s2_bi, s2_Wo, s2_bo, BUF1, BUF0, p2);  // h2=BUF0
  // layer 3
  layer(BUF0, GHID, W3, b3, g3, be3, s3_Wi, s3_bi, s3_Wo, s3_bo, BUF0, BUF1, p3);  // h3=BUF1

  // head
  k_build_hg<<<cdiv(GG * (3 * GHID + GEX), 256), 256, 0, stream>>>(p1, p2, p3, cnt, eF, hg);
  gemm(hg, fc1_W, hid2, GG, 3 * GHID + GEX, 0);
  k_relu_bias<<<cdiv(GG * GHID, 256), 256, 0, stream>>>(hid2, fc1_b);
  k_fc3<<<cdiv(GG * GOUT, 256), 256, 0, stream>>>(hid2, fc3_W, fc3_b, (float*)d_out);
}